// GraphTransformer_29858612642347
// MI455X (gfx1250) — compile-verified
//
#include <hip/hip_runtime.h>
#include <hip/hip_bf16.h>

// ---------------- constants (match reference) ----------------
#define BSZ 2
#define NN  256
#define DX  256
#define DE  64
#define DY  64
#define DFFX 1024
#define DFFE 128
#define DFFY 2048
#define IN_X 64
#define IN_E 16
#define IN_Y 16
#define HMX 128
#define HME 64
#define HMY 64
#define OUT_X 64
#define NEGV (-1000000000.0f)
#define EPSV 1e-5f

#define NROWS_X (BSZ*NN)          // 512
#define NROWS_E (BSZ*NN*NN)       // 131072

typedef float v2f __attribute__((ext_vector_type(2)));
typedef float v8f __attribute__((ext_vector_type(8)));

// ---------------------------------------------------------------------------
// Generic fp32 WMMA GEMM: C[M x N] = act( (PM? add+(mul+1)*A : A) @ W + bias
//                                         (+ C if beta) ) * rowmask
// One wave computes one 16x16 tile with V_WMMA_F32_16X16X4_F32.
// No in-loop predication: out-of-range A rows are address-clamped; their
// garbage only reaches accumulator rows that are never stored.
// ---------------------------------------------------------------------------
template <bool PM>
__global__ __launch_bounds__(256) void wmma_gemm_kernel(
    const float* __restrict__ A, int lda,
    const float* __restrict__ W,          // K x N row-major
    const float* __restrict__ bias,       // N (may be null)
    float* __restrict__ C, int ldc,
    int M, int K, int Nout,
    int act, int beta,
    const float* __restrict__ rowmask,    // M (may be null)
    const float* __restrict__ pm_add,     // BS x K (PM only)
    const float* __restrict__ pm_mul,     // BS x K (PM only)
    int bshift)
{
    int tilesN = Nout >> 4;
    int tilesM = (M + 15) >> 4;
    int wave   = blockIdx.x * (blockDim.x >> 5) + (threadIdx.x >> 5);
    if (wave >= tilesM * tilesN) return;
    int tm = wave / tilesN;
    int tn = wave - tm * tilesN;

    int lane = threadIdx.x & 31;
    int nL   = lane & 15;          // A: row-in-tile ; B/C: col-in-tile
    int hi   = lane >> 4;          // lane half
    int koff = hi * 2;             // K offset of this lane's 2 A/B VGPRs

    int rowA = tm * 16 + nL;
    int rowC = rowA < M ? rowA : (M - 1);   // address clamp (no predication)
    int col  = tn * 16 + nL;

    const float* Ap = A + (size_t)rowC * lda + koff;
    const float* Wp = W + (size_t)koff * Nout + col;
    const float* pma = nullptr;
    const float* pmm = nullptr;
    if (PM) {
        int bIdx = rowC >> bshift;
        pma = pm_add + (size_t)bIdx * K + koff;
        pmm = pm_mul + (size_t)bIdx * K + koff;
    }
    const size_t wstep = (size_t)4 * Nout;

    v8f acc = {};
    for (int k = 0; k < K; k += 4) {
        float a0 = Ap[0];
        float a1 = Ap[1];
        if (PM) {
            a0 = pma[0] + (pmm[0] + 1.f) * a0;
            a1 = pma[1] + (pmm[1] + 1.f) * a1;
            pma += 4; pmm += 4;
        }
        v2f av = { a0, a1 };
        v2f bv = { Wp[0], Wp[Nout] };
        __builtin_prefetch(Ap + 4, 0, 1);
        __builtin_prefetch(Wp + wstep, 0, 1);
        acc = __builtin_amdgcn_wmma_f32_16x16x4_f32(
                  false, av, false, bv, (short)0, acc, false, false);
        Ap += 4;
        Wp += wstep;
    }

    float bval = bias ? bias[col] : 0.f;
    for (int vi = 0; vi < 8; ++vi) {
        int r = tm * 16 + vi + 8 * hi;       // C layout: lane half selects rows 0-7 / 8-15
        if (r < M) {
            float o = acc[vi] + bval;
            float* cp = &C[(size_t)r * ldc + col];
            if (beta) o += *cp;
            if (act)  o = fmaxf(o, 0.f);
            if (rowmask) o *= rowmask[r];
            *cp = o;
        }
    }
}

// ---------------------------------------------------------------------------
// Fused Y = (Q*K/sqrt(DF)) * ((E@Wmul+bmul)*emask + 1) + (E@Wadd+badd)*emask
// M = BS*N*N rows of E (K=64), Nout = 256.  Two WMMA accumulators share A.
// ---------------------------------------------------------------------------
__global__ __launch_bounds__(256) void y_attn_kernel(
    const float* __restrict__ E,
    const float* __restrict__ Wmul, const float* __restrict__ bmul,
    const float* __restrict__ Wadd, const float* __restrict__ badd,
    const float* __restrict__ Qb,   const float* __restrict__ Kb,
    const float* __restrict__ emask,
    float* __restrict__ Y)
{
    int wave = blockIdx.x * 8 + (threadIdx.x >> 5);   // 131072 waves exactly
    int tm = wave >> 4;            // 8192 M-tiles
    int tn = wave & 15;            // 16  N-tiles
    int lane = threadIdx.x & 31;
    int nL = lane & 15, hi = lane >> 4, koff = hi * 2;

    int rowA = tm * 16 + nL;
    int col  = tn * 16 + nL;

    const float* Ap = E + (size_t)rowA * DE + koff;
    const float* Wm = Wmul + (size_t)koff * DX + col;
    const float* Wa = Wadd + (size_t)koff * DX + col;
    const size_t wstep = (size_t)4 * DX;

    v8f acc1 = {}, acc2 = {};
    for (int k = 0; k < DE; k += 4) {
        v2f av = { Ap[0], Ap[1] };
        v2f bm = { Wm[0], Wm[DX] };
        v2f ba = { Wa[0], Wa[DX] };
        __builtin_prefetch(Ap + 4, 0, 1);
        acc1 = __builtin_amdgcn_wmma_f32_16x16x4_f32(false, av, false, bm, (short)0, acc1, false, false);
        acc2 = __builtin_amdgcn_wmma_f32_16x16x4_f32(false, av, false, ba, (short)0, acc2, false, false);
        Ap += 4; Wm += wstep; Wa += wstep;
    }

    const float inv_sqrt_df = 0.17677669529663687f;   // 1/sqrt(32)
    float bmv = bmul[col], bav = badd[col];
    for (int vi = 0; vi < 8; ++vi) {
        int r  = tm * 16 + vi + 8 * hi;
        int b  = r >> 16, q = (r >> 8) & 255, kk = r & 255;
        float em = emask[r];
        float e1 = (acc1[vi] + bmv) * em;
        float e2 = (acc2[vi] + bav) * em;
        float y0 = Qb[(size_t)(b * NN + q) * DX + col] *
                   Kb[(size_t)(b * NN + kk) * DX + col] * inv_sqrt_df;
        Y[(size_t)r * DX + col] = y0 * (e1 + 1.f) + e2;
    }
}

// ---------------------------------------------------------------------------
// Streaming masked softmax over k + weighted V.  block=(b,q), thread=feature.
// wV[b,q,f] = sum_k softmax_k( nm[b,k]?Y:NEG ) * V[b,k,f]
// ---------------------------------------------------------------------------
__global__ __launch_bounds__(256) void attn_wv_kernel(
    const float* __restrict__ Y, const float* __restrict__ V,
    const float* __restrict__ nm, float* __restrict__ wV)
{
    int bq = blockIdx.x;           // 0..511
    int b  = bq >> 8;
    int f  = threadIdx.x;          // 0..255
    const float* yrow  = Y + (size_t)bq * NN * DX + f;
    const float* vbase = V + (size_t)b * NN * DX + f;
    const float* nmb   = nm + b * NN;

    float best = -3.4e38f;
    for (int k = 0; k < NN; ++k) {
        float v = (nmb[k] > 0.f) ? yrow[(size_t)k * DX] : NEGV;
        best = fmaxf(best, v);
    }
    float s = 0.f;
    for (int k = 0; k < NN; ++k) {
        float v = (nmb[k] > 0.f) ? yrow[(size_t)k * DX] : NEGV;
        s += expf(v - best);
    }
    float inv = 1.f / s;
    float acc = 0.f;
    for (int k = 0; k < NN; ++k) {
        float v = (nmb[k] > 0.f) ? yrow[(size_t)k * DX] : NEGV;
        acc += expf(v - best) * inv * vbase[(size_t)k * DX];
    }
    wV[(size_t)bq * DX + f] = acc;
}

// ---------------------------------------------------------------------------
// out = LN(a + b) * g + beta   (row-local; in-place on `a` is safe)
// ---------------------------------------------------------------------------
__global__ void add_ln_kernel(const float* __restrict__ a, const float* __restrict__ bsrc,
                              const float* __restrict__ g, const float* __restrict__ beta,
                              float* __restrict__ out, int d)
{
    __shared__ float s1[256], s2[256];
    size_t row = blockIdx.x;
    int t = threadIdx.x;
    float x = a[row * d + t] + bsrc[row * d + t];
    s1[t] = x; s2[t] = x * x;
    __syncthreads();
    for (int o = d >> 1; o > 0; o >>= 1) {
        if (t < o) { s1[t] += s1[t + o]; s2[t] += s2[t + o]; }
        __syncthreads();
    }
    float m   = s1[0] / (float)d;
    float var = s2[0] / (float)d - m * m;
    out[row * d + t] = (x - m) * rsqrtf(var + EPSV) * g[t] + beta[t];
}

// Eh = 0.5*(Ein + Ein^T(1,2)) * emask
__global__ void sym_kernel(const float* __restrict__ Ein, const float* __restrict__ nm,
                           float* __restrict__ Eout)
{
    size_t i = (size_t)blockIdx.x * blockDim.x + threadIdx.x;  // 8388608
    int f = i & 63;
    size_t r = i >> 6;
    int kk = r & 255, q = (r >> 8) & 255, b = r >> 16;
    float atr = Ein[(((size_t)b * NN + kk) * NN + q) * DE + f];
    Eout[i] = 0.5f * (Ein[i] + atr) * nm[b * NN + q] * nm[b * NN + kk];
}

__global__ void emask_kernel(const float* __restrict__ nm, float* __restrict__ em)
{
    int i = blockIdx.x * 256 + threadIdx.x;   // 131072
    int b = i >> 16, q = (i >> 8) & 255, kk = i & 255;
    em[i] = nm[b * NN + q] * nm[b * NN + kk];
}

// xtoy: concat[mean,min,max,var] over nodes -> (BS, 4*DX)
__global__ void xtoy_kernel(const float* __restrict__ X, const float* __restrict__ nm,
                            float* __restrict__ outc)
{
    int b = blockIdx.x, f = threadIdx.x;      // 256 threads
    float s = 0.f, mi = 3.4e38f, ma = -3.4e38f, denom = 0.f;
    for (int n = 0; n < NN; ++n) {
        float x = X[((size_t)b * NN + n) * DX + f];
        float msk = nm[b * NN + n];
        s += x; denom += msk;
        mi = fminf(mi, x + 100000.f * (1.f - msk));
        ma = fmaxf(ma, x - 100000.f * (1.f - msk));
    }
    float m = s / denom;
    float var = 0.f;
    for (int n = 0; n < NN; ++n) {
        float x = X[((size_t)b * NN + n) * DX + f];
        float d = (x - m) * nm[b * NN + n];
        var += d * d;
    }
    outc[b * (4 * DX) + f]            = m;
    outc[b * (4 * DX) + DX + f]       = mi;
    outc[b * (4 * DX) + 2 * DX + f]   = ma;
    outc[b * (4 * DX) + 3 * DX + f]   = var / denom;
}

// etoy stage 1: per (b,q) partial sum/min/max over k
__global__ void etoy_part1(const float* __restrict__ E, const float* __restrict__ nm,
                           float* __restrict__ psum, float* __restrict__ pmin,
                           float* __restrict__ pmax)
{
    int bq = blockIdx.x, b = bq >> 8, q = bq & 255, f = threadIdx.x;  // 64 threads
    float mq = nm[b * NN + q];
    float s = 0.f, mi = 3.4e38f, ma = -3.4e38f;
    for (int k = 0; k < NN; ++k) {
        float x = E[((size_t)bq * NN + k) * DE + f];
        float im = 1.f - mq * nm[b * NN + k];
        s += x;
        mi = fminf(mi, x + 100000.f * im);
        ma = fmaxf(ma, x - 100000.f * im);
    }
    psum[bq * DE + f] = s; pmin[bq * DE + f] = mi; pmax[bq * DE + f] = ma;
}

// etoy stage 2: finalize mean/min/max -> concatE[0..191]
__global__ void etoy_part2(const float* __restrict__ psum, const float* __restrict__ pmin,
                           const float* __restrict__ pmax, const float* __restrict__ nm,
                           float* __restrict__ outc)
{
    int b = blockIdx.x, f = threadIdx.x;      // 64 threads
    float s = 0.f, mi = 3.4e38f, ma = -3.4e38f;
    for (int q = 0; q < NN; ++q) {
        int i = (b * NN + q) * DE + f;
        s += psum[i]; mi = fminf(mi, pmin[i]); ma = fmaxf(ma, pmax[i]);
    }
    float cnt = 0.f;
    for (int n = 0; n < NN; ++n) cnt += nm[b * NN + n];
    float denom = cnt * cnt;
    outc[b * (4 * DE) + f]          = s / denom;
    outc[b * (4 * DE) + DE + f]     = mi;
    outc[b * (4 * DE) + 2 * DE + f] = ma;
}

// etoy stage 3: per (b,q) partial variance using mean from concatE
__global__ void etoy_var1(const float* __restrict__ E, const float* __restrict__ nm,
                          const float* __restrict__ outc, float* __restrict__ pvar)
{
    int bq = blockIdx.x, b = bq >> 8, q = bq & 255, f = threadIdx.x;
    float m = outc[b * (4 * DE) + f];
    float mq = nm[b * NN + q];
    float s = 0.f;
    for (int k = 0; k < NN; ++k) {
        float d = (E[((size_t)bq * NN + k) * DE + f] - m) * (mq * nm[b * NN + k]);
        s += d * d;
    }
    pvar[bq * DE + f] = s;
}

// etoy stage 4: finalize variance -> concatE[192..255]
__global__ void etoy_var2(const float* __restrict__ pvar, const float* __restrict__ nm,
                          float* __restrict__ outc)
{
    int b = blockIdx.x, f = threadIdx.x;
    float s = 0.f;
    for (int q = 0; q < NN; ++q) s += pvar[(b * NN + q) * DE + f];
    float cnt = 0.f;
    for (int n = 0; n < NN; ++n) cnt += nm[b * NN + n];
    outc[b * (4 * DE) + 3 * DE + f] = s / (cnt * cnt);
}

// ---------------------------------------------------------------------------
// Host side
// ---------------------------------------------------------------------------
struct LinP { const float* b; const float* w; };
struct LNP  { const float* b; const float* g; };
struct LayerP {
    LinP e_add, e_mul, e_out, e_y, k, q, v, x_out, x_y,
         y_e_add, y_e_mul, y_out1, y_out2, y_x_add, y_x_mul, y_y;
    LinP linE1, linE2, linX1, linX2, lin_y1, lin_y2;
    LNP  normE1, normE2, normX1, normX2, norm_y1, norm_y2;
};

static void gemm(hipStream_t s, const float* A, int lda, const float* W,
                 const float* bias, float* C, int ldc, int M, int K, int Nout,
                 int act, int beta, const float* rowmask = nullptr,
                 const float* pma = nullptr, const float* pmm = nullptr, int bshift = 0)
{
    int tiles  = ((M + 15) / 16) * (Nout / 16);
    int blocks = (tiles + 7) / 8;
    if (pma)
        wmma_gemm_kernel<true><<<blocks, 256, 0, s>>>(A, lda, W, bias, C, ldc, M, K, Nout,
                                                      act, beta, rowmask, pma, pmm, bshift);
    else
        wmma_gemm_kernel<false><<<blocks, 256, 0, s>>>(A, lda, W, bias, C, ldc, M, K, Nout,
                                                       act, beta, rowmask, nullptr, nullptr, 0);
}

extern "C" void kernel_launch(void* const* d_in, const int* in_sizes, int n_in,
                              void* d_out, int out_size, void* d_ws, size_t ws_size,
                              hipStream_t stream)
{
    const float* Xin = (const float*)d_in[0];   // (2,256,64)
    const float* Ein = (const float*)d_in[1];   // (2,256,256,16)
    const float* yin = (const float*)d_in[2];   // (2,16)
    const float* nm  = (const float*)d_in[3];   // (2,256)

    // ---- params in JAX pytree order (sorted dict keys; 'b' before 'w') ----
    const float* const* P = (const float* const*)(d_in + 4);
    int pi = 0;
    auto nextLin = [&]() { LinP l; l.b = P[pi++]; l.w = P[pi++]; return l; };
    auto nextLN  = [&]() { LNP  l; l.b = P[pi++]; l.g = P[pi++]; return l; };

    LayerP Ls[2];
    for (int i = 0; i < 2; ++i) {
        LayerP& L = Ls[i];
        L.e_add = nextLin(); L.e_mul = nextLin(); L.e_out = nextLin(); L.e_y = nextLin();
        L.k = nextLin(); L.q = nextLin(); L.v = nextLin(); L.x_out = nextLin();
        L.x_y = nextLin();
        L.y_e_add = nextLin(); L.y_e_mul = nextLin();
        L.y_out1 = nextLin(); L.y_out2 = nextLin();
        L.y_x_add = nextLin(); L.y_x_mul = nextLin(); L.y_y = nextLin();
        L.linE1 = nextLin(); L.linE2 = nextLin();
        L.linX1 = nextLin(); L.linX2 = nextLin();
        L.lin_y1 = nextLin(); L.lin_y2 = nextLin();
        L.normE1 = nextLN(); L.normE2 = nextLN();
        L.normX1 = nextLN(); L.normX2 = nextLN();
        L.norm_y1 = nextLN(); L.norm_y2 = nextLN();
    }
    LinP inE0 = nextLin(), inE1 = nextLin();
    LinP inX0 = nextLin(), inX1 = nextLin();
    LinP iny0 = nextLin(), iny1 = nextLin();
    LinP o0   = nextLin(), o1   = nextLin();

    // ---- workspace layout (floats) ----
    float* ws = (float*)d_ws;
    size_t off = 0;
    auto take = [&](size_t n) { float* p = ws + off; off += n; return p; };
    float* Ybuf   = take((size_t)NROWS_E * DX);     // 33.5M floats (reused as FFE hidden)
    float* Ecur   = take((size_t)NROWS_E * DE);
    float* Etmp   = take((size_t)NROWS_E * DE);
    float* Xcur   = take((size_t)NROWS_X * DX);
    float* Qb     = take((size_t)NROWS_X * DX);
    float* Kb     = take((size_t)NROWS_X * DX);
    float* Vb     = take((size_t)NROWS_X * DX);
    float* wVb    = take((size_t)NROWS_X * DX);
    float* Xtmp   = take((size_t)NROWS_X * DFFX);
    float* Xtmp2  = take((size_t)NROWS_X * DX);
    float* emaskb = take((size_t)NROWS_E);
    float* yh     = take(BSZ * DY);
    float* ytmp   = take(BSZ * DFFY);
    float* ytmp2  = take(BSZ * DY);
    float* ynew   = take(BSZ * DY);
    float* ynew2  = take(BSZ * DY);
    float* ye_add = take(BSZ * DX);
    float* ye_mul = take(BSZ * DX);
    float* yx_add = take(BSZ * DX);
    float* yx_mul = take(BSZ * DX);
    float* concatX = take(BSZ * 4 * DX);
    float* concatE = take(BSZ * 4 * DE);
    float* psum   = take((size_t)BSZ * NN * DE);
    float* pmin   = take((size_t)BSZ * NN * DE);
    float* pmax   = take((size_t)BSZ * NN * DE);
    float* pvar   = take((size_t)BSZ * NN * DE);
    (void)ws_size; (void)in_sizes; (void)n_in; (void)out_size;

    // ---- masks ----
    emask_kernel<<<NROWS_E / 256, 256, 0, stream>>>(nm, emaskb);

    // ---- input MLPs ----
    gemm(stream, Xin, IN_X, inX0.w, inX0.b, Xtmp, HMX, NROWS_X, IN_X, HMX, 1, 0);
    gemm(stream, Xtmp, HMX, inX1.w, inX1.b, Xcur, DX, NROWS_X, HMX, DX, 1, 0, nm);

    gemm(stream, Ein, IN_E, inE0.w, inE0.b, Etmp, HME, NROWS_E, IN_E, HME, 1, 0);
    gemm(stream, Etmp, HME, inE1.w, inE1.b, Ybuf, DE, NROWS_E, HME, DE, 1, 0);
    sym_kernel<<<(NROWS_E * DE) / 256, 256, 0, stream>>>(Ybuf, nm, Ecur);

    gemm(stream, yin, IN_Y, iny0.w, iny0.b, ytmp2, HMY, BSZ, IN_Y, HMY, 1, 0);
    gemm(stream, ytmp2, HMY, iny1.w, iny1.b, yh, DY, BSZ, HMY, DY, 1, 0);

    // ---- transformer layers ----
    for (int li = 0; li < 2; ++li) {
        const LayerP& L = Ls[li];

        // y -> FiLM vectors
        gemm(stream, yh, DY, L.y_e_add.w, L.y_e_add.b, ye_add, DX, BSZ, DY, DX, 0, 0);
        gemm(stream, yh, DY, L.y_e_mul.w, L.y_e_mul.b, ye_mul, DX, BSZ, DY, DX, 0, 0);
        gemm(stream, yh, DY, L.y_x_add.w, L.y_x_add.b, yx_add, DX, BSZ, DY, DX, 0, 0);
        gemm(stream, yh, DY, L.y_x_mul.w, L.y_x_mul.b, yx_mul, DX, BSZ, DY, DX, 0, 0);

        // Q, K, V (masked)
        gemm(stream, Xcur, DX, L.q.w, L.q.b, Qb, DX, NROWS_X, DX, DX, 0, 0, nm);
        gemm(stream, Xcur, DX, L.k.w, L.k.b, Kb, DX, NROWS_X, DX, DX, 0, 0, nm);
        gemm(stream, Xcur, DX, L.v.w, L.v.b, Vb, DX, NROWS_X, DX, DX, 0, 0, nm);

        // fused Y = (QK/sqrt(df))*(E1+1)+E2
        y_attn_kernel<<<(NROWS_E / 16) * (DX / 16) / 8, 256, 0, stream>>>(
            Ecur, L.e_mul.w, L.e_mul.b, L.e_add.w, L.e_add.b, Qb, Kb, emaskb, Ybuf);

        // newE = lin(e_out, ye1 + (ye2+1)*Y) * emask
        gemm(stream, Ybuf, DX, L.e_out.w, L.e_out.b, Etmp, DE,
             NROWS_E, DX, DE, 0, 0, emaskb, ye_add, ye_mul, 16);

        // attention softmax + weighted V
        attn_wv_kernel<<<NROWS_X, 256, 0, stream>>>(Ybuf, Vb, nm, wVb);

        // newX = lin(x_out, yx1 + (yx2+1)*wV) * xmask
        gemm(stream, wVb, DX, L.x_out.w, L.x_out.b, Xtmp2, DX,
             NROWS_X, DX, DX, 0, 0, nm, yx_add, yx_mul, 8);

        // new_y = y_y(yh) + x_y(xtoy) + e_y(etoy)
        gemm(stream, yh, DY, L.y_y.w, L.y_y.b, ynew, DY, BSZ, DY, DY, 0, 0);
        xtoy_kernel<<<BSZ, DX, 0, stream>>>(Xcur, nm, concatX);
        gemm(stream, concatX, 4 * DX, L.x_y.w, L.x_y.b, ynew, DY, BSZ, 4 * DX, DY, 0, 1);
        etoy_part1<<<BSZ * NN, DE, 0, stream>>>(Ecur, nm, psum, pmin, pmax);
        etoy_part2<<<BSZ, DE, 0, stream>>>(psum, pmin, pmax, nm, concatE);
        etoy_var1<<<BSZ * NN, DE, 0, stream>>>(Ecur, nm, concatE, pvar);
        etoy_var2<<<BSZ, DE, 0, stream>>>(pvar, nm, concatE);
        gemm(stream, concatE, 4 * DE, L.e_y.w, L.e_y.b, ynew, DY, BSZ, 4 * DE, DY, 0, 1);
        gemm(stream, ynew, DY, L.y_out1.w, L.y_out1.b, ytmp2, DY, BSZ, DY, DY, 1, 0);
        gemm(stream, ytmp2, DY, L.y_out2.w, L.y_out2.b, ynew2, DY, BSZ, DY, DY, 0, 0);

        // residual + LN (block 1)
        add_ln_kernel<<<NROWS_X, DX, 0, stream>>>(Xcur, Xtmp2, L.normX1.g, L.normX1.b, Xcur, DX);
        add_ln_kernel<<<NROWS_E, DE, 0, stream>>>(Ecur, Etmp, L.normE1.g, L.normE1.b, Ecur, DE);
        add_ln_kernel<<<BSZ, DY, 0, stream>>>(yh, ynew2, L.norm_y1.g, L.norm_y1.b, yh, DY);

        // FFX
        gemm(stream, Xcur, DX, L.linX1.w, L.linX1.b, Xtmp, DFFX, NROWS_X, DX, DFFX, 1, 0);
        gemm(stream, Xtmp, DFFX, L.linX2.w, L.linX2.b, Xtmp2, DX, NROWS_X, DFFX, DX, 0, 0);
        add_ln_kernel<<<NROWS_X, DX, 0, stream>>>(Xcur, Xtmp2, L.normX2.g, L.normX2.b, Xcur, DX);

        // FFE (hidden reuses dead Ybuf)
        gemm(stream, Ecur, DE, L.linE1.w, L.linE1.b, Ybuf, DFFE, NROWS_E, DE, DFFE, 1, 0);
        gemm(stream, Ybuf, DFFE, L.linE2.w, L.linE2.b, Etmp, DE, NROWS_E, DFFE, DE, 0, 0);
        add_ln_kernel<<<NROWS_E, DE, 0, stream>>>(Ecur, Etmp, L.normE2.g, L.normE2.b, Ecur, DE);

        // FFy
        gemm(stream, yh, DY, L.lin_y1.w, L.lin_y1.b, ytmp, DFFY, BSZ, DY, DFFY, 1, 0);
        gemm(stream, ytmp, DFFY, L.lin_y2.w, L.lin_y2.b, ynew2, DY, BSZ, DFFY, DY, 0, 0);
        add_ln_kernel<<<BSZ, DY, 0, stream>>>(yh, ynew2, L.norm_y2.g, L.norm_y2.b, yh, DY);
    }

    // ---- output head: out = (lin(o1, relu(lin(o0, X))) + X_in) * xmask ----
    gemm(stream, Xcur, DX, o0.w, o0.b, Xtmp, HMX, NROWS_X, DX, HMX, 1, 0);
    hipMemcpyAsync(d_out, (const void*)Xin, (size_t)NROWS_X * OUT_X * sizeof(float),
                   hipMemcpyDeviceToDevice, stream);
    gemm(stream, Xtmp, HMX, o1.w, o1.b, (float*)d_out, OUT_X,
         NROWS_X, HMX, OUT_X, 0, /*beta=*/1, nm);
}